// ContinuousConv_82566451298747
// MI455X (gfx1250) — compile-verified
//
#include <hip/hip_runtime.h>

// ---------------------------------------------------------------------------
// ContinuousConv for MI455X (gfx1250), fp32 end-to-end via V_WMMA_F32_16X16X4_F32.
// One workgroup (256 threads = 8 wave32) handles a tile of 16 output points.
//   Phase 1: gather + trilinear hat weights + stage neighbor features to LDS
//   Phase 2: conflict-free accumulation of feat_grid[16][27][64] in LDS
//   Phase 3: C[16x128] = A[16x1728](LDS) * B[1728x128](kernel, L2-resident)
//            as 8 waves x 432 fp32 WMMA steps (one 16-col N-tile per wave)
// A-fragment LDS stride is padded so stride % 64 banks == 4: the wave32
// ds_load_b64 then touches banks {4*mrow + 2*khalf + 0,1} = all 64 banks
// exactly once -> conflict-free A reads.
// ---------------------------------------------------------------------------

typedef __attribute__((ext_vector_type(2))) float v2f;
typedef __attribute__((ext_vector_type(8))) float v8f;

#define K_NBR   32
#define CIN     64
#define COUT    128
#define CELLS   27
#define TILE_M  16
#define PC      (CELLS * CIN)        // 1728 = GEMM inner dim
#define PC_PAD  (PC + 4)             // 1732: stride % 64 == 4 -> conflict-free b64
#define NPAIR   (TILE_M * K_NBR)     // 512 (point, neighbor) pairs per tile
#define HSTR    10                   // 9 hat values + 1 pad
#define EPS_W   1e-8f

// LDS layout (in floats)
#define OFF_FFEAT  0
#define OFF_FGRID  (NPAIR * CIN)                   // 32768
#define OFF_HBUF   (OFF_FGRID + TILE_M * PC_PAD)   // 60480
#define OFF_NIDX   (OFF_HBUF + NPAIR * HSTR)       // 65600
#define LDS_FLOATS (OFF_NIDX + NPAIR)              // 66112 floats = 264448 B < 320 KB
static_assert(LDS_FLOATS * 4 <= 320 * 1024, "LDS budget exceeded");

__device__ __forceinline__ float hat_w(float g, float node) {
    return fmaxf(0.0f, 1.0f - fabsf(g - node));
}

__global__ void ContinuousConv_82566451298747_kernel(
    const float* __restrict__ ifeat,   // [N,64]
    const float* __restrict__ ipos,    // [N,3]
    const float* __restrict__ opos,    // [M,3]
    const float* __restrict__ extents, // [1]
    const int*   __restrict__ nbridx,  // [M,32]
    const float* __restrict__ kmat,    // [27,64,128] row-major = [1728,128]
    const float* __restrict__ bias,    // [128]
    float*       __restrict__ out,     // [M,128]
    int M)
{
    extern __shared__ float smem[];
    float* ffeat = smem + OFF_FFEAT;   // [512][64] gathered neighbor features
    float* fgrid = smem + OFF_FGRID;   // [16][1732] interpolated feature grid (A matrix)
    float* hbuf  = smem + OFF_HBUF;    // [512][10] per-axis hat weights
    int*   nidx  = (int*)(smem + OFF_NIDX); // [512] neighbor indices

    const int tid = threadIdx.x;
    const int m0  = blockIdx.x * TILE_M;

    // ---------------- Phase 1a: weights + neighbor indices --------------------
    const float s2 = 2.0f / extents[0];
    for (int q = tid; q < NPAIR; q += 256) {
        const int p = q >> 5;
        const int k = q & 31;
        int m = m0 + p; if (m >= M) m = M - 1;   // clamp (keeps EXEC full)
        const int nb = nbridx[m * K_NBR + k];
        nidx[q] = nb;
        const float ox = opos[m * 3 + 0];
        const float oy = opos[m * 3 + 1];
        const float oz = opos[m * 3 + 2];
        const float rx = (ipos[nb * 3 + 0] - ox) * s2;
        const float ry = (ipos[nb * 3 + 1] - oy) * s2;
        const float rz = (ipos[nb * 3 + 2] - oz) * s2;
        // ball -> cube radial map
        const float nrm = sqrtf(rx * rx + ry * ry + rz * rz);
        const float am  = fmaxf(fabsf(rx), fmaxf(fabsf(ry), fabsf(rz)));
        const float sc  = (am > EPS_W) ? (nrm / am) : 0.0f;
        // align_corners grid coords in [0,2] (sizes-1 == 2 on every axis)
        const float gx = fminf(fmaxf(rx * sc + 1.0f, 0.0f), 2.0f);
        const float gy = fminf(fmaxf(ry * sc + 1.0f, 0.0f), 2.0f);
        const float gz = fminf(fmaxf(rz * sc + 1.0f, 0.0f), 2.0f);
        float* h = &hbuf[q * HSTR];
        h[0] = hat_w(gx, 0.0f); h[1] = hat_w(gx, 1.0f); h[2] = hat_w(gx, 2.0f);
        h[3] = hat_w(gy, 0.0f); h[4] = hat_w(gy, 1.0f); h[5] = hat_w(gy, 2.0f);
        h[6] = hat_w(gz, 0.0f); h[7] = hat_w(gz, 1.0f); h[8] = hat_w(gz, 2.0f);
    }
    __syncthreads();

    // ---------------- Phase 1b: stage neighbor features (coalesced) -----------
    for (int t = tid; t < NPAIR * (CIN / 4); t += 256) {
        const int q  = t >> 4;
        const int c4 = t & 15;
        const float4 f = reinterpret_cast<const float4*>(ifeat)[nidx[q] * (CIN / 4) + c4];
        reinterpret_cast<float4*>(ffeat)[q * (CIN / 4) + c4] = f;
    }
    __syncthreads();

    // ---------------- Phase 2: feat_grid accumulation (no atomics) ------------
    {
        const int p  = tid >> 4;   // output point within tile
        const int c4 = tid & 15;   // 4-channel group
        const float invK = 1.0f / (float)K_NBR;
        for (int iz = 0; iz < 3; ++iz) {
            float ax[9], ay[9], az9[9], aw[9];
#pragma unroll
            for (int j = 0; j < 9; ++j) { ax[j] = ay[j] = az9[j] = aw[j] = 0.0f; }
            for (int k = 0; k < K_NBR; ++k) {
                const int q = p * K_NBR + k;
                const float* h = &hbuf[q * HSTR];
                const float hz = h[6 + iz];
                const float hx[3] = { h[0], h[1], h[2] };
                const float wy[3] = { hz * h[3], hz * h[4], hz * h[5] };
                const float4 f = reinterpret_cast<const float4*>(ffeat)[q * (CIN / 4) + c4];
#pragma unroll
                for (int iy = 0; iy < 3; ++iy) {
#pragma unroll
                    for (int ix = 0; ix < 3; ++ix) {
                        const float w = wy[iy] * hx[ix];
                        const int j = iy * 3 + ix;
                        ax[j]  += w * f.x;
                        ay[j]  += w * f.y;
                        az9[j] += w * f.z;
                        aw[j]  += w * f.w;
                    }
                }
            }
#pragma unroll
            for (int iy = 0; iy < 3; ++iy) {
#pragma unroll
                for (int ix = 0; ix < 3; ++ix) {
                    const int cell = (iz * 3 + iy) * 3 + ix;
                    const int j = iy * 3 + ix;
                    float4 v;
                    v.x = ax[j] * invK; v.y = ay[j] * invK;
                    v.z = az9[j] * invK; v.w = aw[j] * invK;
                    *reinterpret_cast<float4*>(&fgrid[p * PC_PAD + cell * CIN + c4 * 4]) = v;
                }
            }
        }
    }
    __syncthreads();

    // ---------------- Phase 3: WMMA GEMM  C[16x128] = A[16x1728] * B[1728x128] -
    {
        const int wv    = tid >> 5;    // wave id -> 16-column N tile
        const int lane  = tid & 31;
        const int mrow  = lane & 15;   // A: M across lanes 0..15 / 16..31
        const int khalf = lane >> 4;   // A: K {0,1} (lanes 0-15) vs K {2,3} (lanes 16-31)
        // A fragment: lane holds A[mrow][kk*4 + khalf*2 + {0,1}]
        const float* Abase = fgrid + mrow * PC_PAD + khalf * 2;
        // B fragment: lane holds B[kk*4 + khalf*2 + {0,1}][wv*16 + mrow]
        const float* Bbase = kmat + (khalf * 2) * COUT + wv * 16 + mrow;

        v8f acc = {0.f, 0.f, 0.f, 0.f, 0.f, 0.f, 0.f, 0.f};
#pragma unroll 8
        for (int kk = 0; kk < PC / 4; ++kk) {
            // prefetch the kernel-matrix stream ~8 K-steps (16 KB) ahead
            if ((kk & 7) == 0) {
                __builtin_prefetch(Bbase + 8 * 4 * COUT, 0, 1);
            }
            v2f a = *reinterpret_cast<const v2f*>(Abase);   // ds_load_b64, bank-conflict-free
            v2f b;
            b.x = Bbase[0];
            b.y = Bbase[COUT];
            acc = __builtin_amdgcn_wmma_f32_16x16x4_f32(
                false, a, false, b, (short)0, acc, false, false);
            Abase += 4;
            Bbase += 4 * COUT;
        }

        const float bv = bias[wv * 16 + mrow];
        // C layout: VGPR r -> M = r (lanes 0-15) / r+8 (lanes 16-31), N = lane&15
        float* outp = out + (size_t)(m0 + khalf * 8) * COUT + wv * 16 + mrow;
        if (m0 + TILE_M <= M) {
            // full tile: branchless, offsets fold into store immediates
#pragma unroll
            for (int r = 0; r < 8; ++r) {
                outp[(size_t)r * COUT] = acc[r] + bv;
            }
        } else {
            // ragged last tile (not hit for M % 16 == 0)
#pragma unroll
            for (int r = 0; r < 8; ++r) {
                const int m = m0 + khalf * 8 + r;
                if (m < M) outp[(size_t)r * COUT] = acc[r] + bv;
            }
        }
    }
}

extern "C" void kernel_launch(void* const* d_in, const int* in_sizes, int n_in,
                              void* d_out, int out_size, void* d_ws, size_t ws_size,
                              hipStream_t stream) {
    const float* ifeat   = (const float*)d_in[0];
    const float* ipos    = (const float*)d_in[1];
    const float* opos    = (const float*)d_in[2];
    const float* extents = (const float*)d_in[3];
    const int*   nbridx  = (const int*)d_in[4];
    const float* kmat    = (const float*)d_in[5];
    const float* bias    = (const float*)d_in[6];
    float* out = (float*)d_out;

    const int M = in_sizes[2] / 3;                  // out_positions is [M,3]
    const int nblocks = (M + TILE_M - 1) / TILE_M;  // 3125 for M=50000
    const size_t shbytes = (size_t)LDS_FLOATS * sizeof(float);

    hipLaunchKernelGGL(ContinuousConv_82566451298747_kernel,
                       dim3(nblocks), dim3(256), shbytes, stream,
                       ifeat, ipos, opos, extents, nbridx, kmat, bias, out, M);
}